// FeatureTargetStatsEncoder_12043088298469
// MI455X (gfx1250) — compile-verified
//
#include <hip/hip_runtime.h>
#include <hip/hip_bf16.h>
#include <math.h>

// ---------------- problem constants (from reference) ----------------
#define NB   64      // batch
#define NN   4096    // samples per column
#define NF   256     // features
#define NCLS 10
#define HID  64
#define ODIM 128
#define EPSV 1e-6f

typedef float v2f __attribute__((ext_vector_type(2)));
typedef float v8f __attribute__((ext_vector_type(8)));

// =====================================================================
// Kernel 1: per-(b,f) column statistics -> stats[B*F, 6] in workspace
// One block (256 threads) per column; column cached in LDS for sort.
// =====================================================================
__global__ __launch_bounds__(256) void fts_stats_kernel(
    const float* __restrict__ X,     // [B, N, F]
    const int*   __restrict__ Y,     // [B, N]
    float*       __restrict__ stats) // [B*F, 6]
{
  __shared__ float col[NN];              // cleaned column (16 KB)
  __shared__ float s_sum, s_sumsq, s_sumabs;
  __shared__ int   s_maxbits, s_nan, s_uniq;
  __shared__ float s_csum[NCLS];
  __shared__ float s_ccnt[NCLS];

  const int tid   = threadIdx.x;
  const int colId = blockIdx.x;          // b*NF + f
  const int b     = colId >> 8;          // /NF (NF==256)
  const int f     = colId & (NF - 1);

  if (tid == 0) {
    s_sum = 0.f; s_sumsq = 0.f; s_sumabs = 0.f;
    s_maxbits = 0; s_nan = 0; s_uniq = 0;
  }
  if (tid < NCLS) { s_csum[tid] = 0.f; s_ccnt[tid] = 0.f; }
  __syncthreads();

  const float* xb = X + (size_t)b * NN * NF + f;
  const int*   yb = Y + (size_t)b * NN;

  float lsum = 0.f, lsq = 0.f, labs = 0.f, lmax = 0.f;
  int   lnan = 0;
  float lcs[NCLS], lcc[NCLS];
#pragma unroll
  for (int c = 0; c < NCLS; ++c) { lcs[c] = 0.f; lcc[c] = 0.f; }

  for (int i = tid; i < NN; i += 256) {
    float v   = xb[(size_t)i * NF];
    bool  nan = (v != v);
    lnan += nan ? 1 : 0;
    float cv = nan ? 0.f : v;
    col[i] = cv;
    lsum += cv;
    lsq   = fmaf(cv, cv, lsq);
    float av = fabsf(cv);
    labs += av;
    lmax  = fmaxf(lmax, av);
    int cls = yb[i];
#pragma unroll
    for (int c = 0; c < NCLS; ++c) {
      bool m = (cls == c);
      lcs[c] += m ? cv  : 0.f;
      lcc[c] += m ? 1.f : 0.f;
    }
  }
  atomicAdd(&s_sum,    lsum);
  atomicAdd(&s_sumsq,  lsq);
  atomicAdd(&s_sumabs, labs);
  atomicMax(&s_maxbits, __float_as_int(lmax));   // lmax >= 0 -> int order ok
  atomicAdd(&s_nan,    lnan);
#pragma unroll
  for (int c = 0; c < NCLS; ++c) {
    atomicAdd(&s_csum[c], lcs[c]);
    atomicAdd(&s_ccnt[c], lcc[c]);
  }
  __syncthreads();

  // ---- bitonic sort (ascending) of col[0..NN) ----
  for (int k = 2; k <= NN; k <<= 1) {
    for (int j = k >> 1; j > 0; j >>= 1) {
      for (int i = tid; i < NN; i += 256) {
        int ixj = i ^ j;
        if (ixj > i) {
          float a = col[i];
          float c2 = col[ixj];
          bool  up = ((i & k) == 0);
          bool  sw = up ? (a > c2) : (a < c2);
          if (sw) { col[i] = c2; col[ixj] = a; }
        }
      }
      __syncthreads();
    }
  }

  // ---- unique count via adjacent diffs on sorted column ----
  int lu = 0;
  for (int i = tid; i < NN; i += 256)
    if (i > 0) lu += (col[i] != col[i - 1]) ? 1 : 0;
  atomicAdd(&s_uniq, lu);
  __syncthreads();

  if (tid == 0) {
    const float invN = 1.0f / (float)NN;
    float mean    = s_sum * invN;
    float var     = fmaxf(s_sumsq * invN - mean * mean, 0.f); // biased var
    float meanAbs = s_sumabs * invN;
    float maxAbs  = __int_as_float(s_maxbits);
    float missing = (float)s_nan * invN;
    float uratio  = (1.0f + (float)s_uniq) * invN;

    float bv = 0.f;
#pragma unroll
    for (int c = 0; c < NCLS; ++c) {
      float cnt = s_ccnt[c];
      float cm  = s_csum[c] / fmaxf(cnt, 1.0f);
      float d   = cm - mean;
      bv = fmaf(cnt, d * d, bv);
    }
    bv *= invN;                               // counts.sum == N
    float score = bv / fmaxf(var, EPSV);

    float o[6] = { score, missing, uratio, var, meanAbs, maxAbs };
    size_t base = (size_t)colId * 6;
#pragma unroll
    for (int k2 = 0; k2 < 6; ++k2) {
      float t = o[k2];
      // jnp.nan_to_num(..., nan=0, posinf=0, neginf=0)
      stats[base + k2] = (t == t && fabsf(t) <= 3.4028234e38f) ? t : 0.f;
    }
  }
}

// =====================================================================
// Kernel 2: MLP  out = gelu(stats@W1+b1) @ W2 + b2  using f32 WMMA.
// 8 waves/block; each wave owns 16 rows. Layer-1 results are produced
// directly in the V_WMMA_F32_16X16X4_F32 A-fragment layout.
// =====================================================================
__global__ __launch_bounds__(256) void fts_mlp_kernel(
    const float* __restrict__ stats, // [B*F, 6]
    const float* __restrict__ W1,    // [6, 64]
    const float* __restrict__ B1,    // [64]
    const float* __restrict__ W2,    // [64, 128]
    const float* __restrict__ B2,    // [128]
    float*       __restrict__ out)   // [B*F, 128]
{
  __shared__ float sW2[HID * ODIM];  // 32 KB
  __shared__ float sW1[6 * HID];
  __shared__ float sB1[HID];
  __shared__ float sB2[ODIM];

  const int tid = threadIdx.x;
  for (int i = tid; i < HID * ODIM; i += 256) sW2[i] = W2[i];
  for (int i = tid; i < 6 * HID;    i += 256) sW1[i] = W1[i];
  if (tid < HID)  sB1[tid] = B1[tid];
  if (tid < ODIM) sB2[tid] = B2[tid];
  __syncthreads();

  const int lane  = tid & 31;
  const int wave  = tid >> 5;
  const int mrow  = lane & 15;     // A/C row (and B/C column index) this lane serves
  const int hi    = lane >> 4;     // 0: K%4 in {0,1}; 1: K%4 in {2,3}
  const int rowBase = (blockIdx.x * 8 + wave) * 16;
  const size_t srow = (size_t)(rowBase + mrow);

  // stats row for this lane's matrix row
  const float* sp = stats + srow * 6;
  float s0 = sp[0], s1 = sp[1], s2 = sp[2], s3 = sp[3], s4 = sp[4], s5 = sp[5];

  // h[mrow][j] for the 32 j-columns this lane owns in the A-fragment layout:
  // hreg[2t+u] = h[mrow][4t + 2*hi + u]
  float hreg[32];
#pragma unroll
  for (int t = 0; t < 16; ++t) {
#pragma unroll
    for (int u = 0; u < 2; ++u) {
      int j = 4 * t + 2 * hi + u;
      float a = sB1[j];
      a = fmaf(s0, sW1[0 * HID + j], a);
      a = fmaf(s1, sW1[1 * HID + j], a);
      a = fmaf(s2, sW1[2 * HID + j], a);
      a = fmaf(s3, sW1[3 * HID + j], a);
      a = fmaf(s4, sW1[4 * HID + j], a);
      a = fmaf(s5, sW1[5 * HID + j], a);
      // exact GELU: 0.5*x*(1+erf(x/sqrt(2)))
      hreg[2 * t + u] = 0.5f * a * (1.0f + erff(a * 0.70710678118654752f));
    }
  }

  const int koff = 2 * hi;
  for (int n0 = 0; n0 < 8; ++n0) {
    const int ncol = n0 * 16 + mrow;
    const float bias = sB2[ncol];
    v8f acc = { bias, bias, bias, bias, bias, bias, bias, bias };
#pragma unroll
    for (int t = 0; t < 16; ++t) {
      v2f a;  a.x = hreg[2 * t];     a.y = hreg[2 * t + 1];
      v2f bm; bm.x = sW2[(4 * t + koff) * ODIM + ncol];
              bm.y = sW2[(4 * t + koff + 1) * ODIM + ncol];
      acc = __builtin_amdgcn_wmma_f32_16x16x4_f32(
          /*neg_a=*/false, a, /*neg_b=*/false, bm,
          /*c_mod=*/(short)0, acc, /*reuse_a=*/false, /*reuse_b=*/false);
    }
#pragma unroll
    for (int r = 0; r < 8; ++r) {
      int row = rowBase + r + 8 * hi;   // C layout: lanes>=16 hold M=r+8
      out[(size_t)row * ODIM + ncol] = acc[r];
    }
  }
}

// =====================================================================
extern "C" void kernel_launch(void* const* d_in, const int* in_sizes, int n_in,
                              void* d_out, int out_size, void* d_ws, size_t ws_size,
                              hipStream_t stream) {
  const float* X  = (const float*)d_in[0];
  const int*   Y  = (const int*)  d_in[1];
  const float* W1 = (const float*)d_in[2];
  const float* b1 = (const float*)d_in[3];
  const float* W2 = (const float*)d_in[4];
  const float* b2 = (const float*)d_in[5];
  float* out   = (float*)d_out;
  float* stats = (float*)d_ws;              // [B*F, 6] = 393 KB scratch

  // Kernel 1: one block per (b,f) column
  hipLaunchKernelGGL(fts_stats_kernel, dim3(NB * NF), dim3(256), 0, stream,
                     X, Y, stats);

  // Kernel 2: 8 waves/block * 16 rows/wave -> 128 rows/block
  hipLaunchKernelGGL(fts_mlp_kernel, dim3((NB * NF) / 128), dim3(256), 0, stream,
                     stats, W1, b1, W2, b2, out);
}